// SelfPointerGeneratorOut_66571993088451
// MI455X (gfx1250) — compile-verified
//
#include <hip/hip_runtime.h>
#include <hip/hip_bf16.h>
#include <math.h>

typedef __attribute__((ext_vector_type(2))) float v2f;
typedef __attribute__((ext_vector_type(8))) float v8f;

namespace {
constexpr int BATCH  = 64;
constexpr int INPDIM = 1024;
constexpr int VECDIM = 512;
constexpr int VOCAB  = 50257;
constexpr int CTX    = 512;
constexpr int PREV   = 64;
constexpr int NTILES = (VOCAB + 15) / 16;   // 3142 column tiles of 16
}

// ---------------------------------------------------------------------------
// Kernel 2: logits = x @ W_gen + b_gen  -> d_out  (f32 WMMA, 16x16x4)
// One wave per 16-wide vocab tile; 4 M-tiles (all 64 rows) per wave.
// ---------------------------------------------------------------------------
__global__ void __launch_bounds__(128)
gen_gemm_kernel(const float* __restrict__ x, const float* __restrict__ Wg,
                const float* __restrict__ bg, float* __restrict__ out) {
    const int wave = (blockIdx.x * blockDim.x + threadIdx.x) >> 5;
    if (wave >= NTILES) return;                    // wave-uniform exit
    const int lane = threadIdx.x & 31;
    const int ml   = lane & 15;                    // M (A) / N (B,C) lane index
    const int hi   = lane >> 4;                    // K half selector for A/B
    const int n0   = wave * 16;
    const int nl   = n0 + ml;
    const bool nvalid = nl < VOCAB;
    const int nc   = nvalid ? nl : (VOCAB - 1);    // clamp: garbage cols never stored

    v8f acc0 = {}, acc1 = {}, acc2 = {}, acc3 = {};

    const float* xr0 = x + (size_t)(ml +  0) * INPDIM;
    const float* xr1 = x + (size_t)(ml + 16) * INPDIM;
    const float* xr2 = x + (size_t)(ml + 32) * INPDIM;
    const float* xr3 = x + (size_t)(ml + 48) * INPDIM;

    #pragma unroll 4
    for (int k = 0; k < INPDIM; k += 4) {
        const int kb = k + hi * 2;
        // A 16x4 f32 layout: lanes 0-15 -> K={k,k+1}, lanes 16-31 -> K={k+2,k+3}
        v2f a0 = *(const v2f*)(xr0 + kb);
        v2f a1 = *(const v2f*)(xr1 + kb);
        v2f a2 = *(const v2f*)(xr2 + kb);
        v2f a3 = *(const v2f*)(xr3 + kb);
        // B 4x16 f32 layout (mirrors C/D row split): VGPR0 = rows {k, k+2}, VGPR1 = {k+1, k+3}
        v2f bb;
        bb.x = Wg[(size_t)(kb    ) * VOCAB + nc];
        bb.y = Wg[(size_t)(kb + 1) * VOCAB + nc];

        acc0 = __builtin_amdgcn_wmma_f32_16x16x4_f32(false, a0, false, bb, (short)0, acc0, false, false);
        acc1 = __builtin_amdgcn_wmma_f32_16x16x4_f32(false, a1, false, bb, (short)0, acc1, false, false);
        acc2 = __builtin_amdgcn_wmma_f32_16x16x4_f32(false, a2, false, bb, (short)0, acc2, false, false);
        acc3 = __builtin_amdgcn_wmma_f32_16x16x4_f32(false, a3, false, bb, (short)0, acc3, false, false);
    }

    if (nvalid) {
        const float bias = bg[nl];
        const int moff = hi * 8;                   // C/D: lanes16-31 hold rows M+8
        #pragma unroll
        for (int r = 0; r < 8; ++r) {
            out[(size_t)( 0 + moff + r) * VOCAB + nl] = acc0[r] + bias;
            out[(size_t)(16 + moff + r) * VOCAB + nl] = acc1[r] + bias;
            out[(size_t)(32 + moff + r) * VOCAB + nl] = acc2[r] + bias;
            out[(size_t)(48 + moff + r) * VOCAB + nl] = acc3[r] + bias;
        }
    }
}

// ---------------------------------------------------------------------------
// Kernel 1: mix[b][3] = softmax(tanh(x@W1+b1)@W2 + b2)
// ---------------------------------------------------------------------------
__global__ void __launch_bounds__(128)
gate_kernel(const float* __restrict__ x, const float* __restrict__ W1,
            const float* __restrict__ b1, const float* __restrict__ W2,
            const float* __restrict__ b2, float* __restrict__ mix) {
    const int b = blockIdx.x, t = threadIdx.x;
    __shared__ float s3[3];
    if (t < 3) s3[t] = 0.0f;
    __syncthreads();
    const float* xr = x + (size_t)b * INPDIM;
    float l0 = 0.f, l1 = 0.f, l2 = 0.f;
    for (int j = t; j < VECDIM; j += 128) {
        float h = b1[j];
        for (int k = 0; k < INPDIM; ++k) h += xr[k] * W1[(size_t)k * VECDIM + j];
        h = tanhf(h);
        l0 += h * W2[j * 3 + 0];
        l1 += h * W2[j * 3 + 1];
        l2 += h * W2[j * 3 + 2];
    }
    atomicAdd(&s3[0], l0); atomicAdd(&s3[1], l1); atomicAdd(&s3[2], l2);
    __syncthreads();
    if (t == 0) {
        float a = s3[0] + b2[0], c = s3[1] + b2[1], d = s3[2] + b2[2];
        float m = fmaxf(a, fmaxf(c, d));
        float e0 = __expf(a - m), e1 = __expf(c - m), e2 = __expf(d - m);
        float inv = 1.0f / (e0 + e1 + e2);
        mix[b * 3 + 0] = e0 * inv;
        mix[b * 3 + 1] = e1 * inv;
        mix[b * 3 + 2] = e2 * inv;
    }
}

// ---------------------------------------------------------------------------
// Kernel 3: per-row max and sum(exp(l - max)) over the vocab
// ---------------------------------------------------------------------------
__global__ void __launch_bounds__(256)
rowstat_kernel(const float* __restrict__ logits, float* __restrict__ rowmax,
               float* __restrict__ rowsum) {
    const int b = blockIdx.x, t = threadIdx.x;
    __shared__ float red[256];
    const float* row = logits + (size_t)b * VOCAB;
    float m = -3.4e38f;
    for (int j = t; j < VOCAB; j += 256) m = fmaxf(m, row[j]);
    red[t] = m; __syncthreads();
    for (int s = 128; s > 0; s >>= 1) { if (t < s) red[t] = fmaxf(red[t], red[t + s]); __syncthreads(); }
    m = red[0]; __syncthreads();
    float sum = 0.f;
    for (int j = t; j < VOCAB; j += 256) sum += __expf(row[j] - m);
    red[t] = sum; __syncthreads();
    for (int s = 128; s > 0; s >>= 1) { if (t < s) red[t] += red[t + s]; __syncthreads(); }
    if (t == 0) { rowmax[b] = m; rowsum[b] = red[0]; }
}

// ---------------------------------------------------------------------------
// Kernel 4: out = exp(logit - max)/sum * mix0   (in place over d_out)
// ---------------------------------------------------------------------------
__global__ void __launch_bounds__(256)
norm_kernel(float* __restrict__ out, const float* __restrict__ rowmax,
            const float* __restrict__ rowsum, const float* __restrict__ mix,
            int total) {
    const int i = blockIdx.x * 256 + threadIdx.x;
    if (i >= total) return;
    const int b = i / VOCAB;
    const float scale = mix[b * 3 + 0] / rowsum[b];
    out[i] = __expf(out[i] - rowmax[b]) * scale;
}

// ---------------------------------------------------------------------------
// Kernel 5: scatter copy + self-copy distributions (atomic, per-row block)
// ---------------------------------------------------------------------------
__global__ void __launch_bounds__(128)
scatter_kernel(const float* __restrict__ scores, const float* __restrict__ selfscores,
               const int* __restrict__ ctx_ids, const int* __restrict__ prev_tokens,
               const float* __restrict__ mix, float* __restrict__ out) {
    const int b = blockIdx.x, t = threadIdx.x;
    __shared__ float red[128];
    float* orow = out + (size_t)b * VOCAB;

    // --- copy distribution over ctx_ids (CTX = 512) ---
    const float* sr = scores + (size_t)b * CTX;
    float m = -3.4e38f;
    for (int j = t; j < CTX; j += 128) m = fmaxf(m, sr[j]);
    red[t] = m; __syncthreads();
    for (int s = 64; s > 0; s >>= 1) { if (t < s) red[t] = fmaxf(red[t], red[t + s]); __syncthreads(); }
    m = red[0]; __syncthreads();
    float sum = 0.f;
    for (int j = t; j < CTX; j += 128) sum += __expf(sr[j] - m);
    red[t] = sum; __syncthreads();
    for (int s = 64; s > 0; s >>= 1) { if (t < s) red[t] += red[t + s]; __syncthreads(); }
    const float coef1 = mix[b * 3 + 1] / red[0];
    __syncthreads();
    for (int j = t; j < CTX; j += 128)
        atomicAdd(&orow[ctx_ids[b * CTX + j]], __expf(sr[j] - m) * coef1);

    // --- self-copy distribution over prev tokens (PREV = 64, row stride 65) ---
    const float* ss = selfscores + (size_t)b * PREV;
    float m2 = -3.4e38f;
    for (int j = t; j < PREV; j += 128) m2 = fmaxf(m2, ss[j]);
    red[t] = m2; __syncthreads();
    for (int s = 64; s > 0; s >>= 1) { if (t < s) red[t] = fmaxf(red[t], red[t + s]); __syncthreads(); }
    m2 = red[0]; __syncthreads();
    float sum2 = 0.f;
    for (int j = t; j < PREV; j += 128) sum2 += __expf(ss[j] - m2);
    red[t] = sum2; __syncthreads();
    for (int s = 64; s > 0; s >>= 1) { if (t < s) red[t] += red[t + s]; __syncthreads(); }
    const float coef2 = mix[b * 3 + 2] / red[0];
    __syncthreads();
    for (int j = t; j < PREV; j += 128)
        atomicAdd(&orow[prev_tokens[b * (PREV + 1) + j]], __expf(ss[j] - m2) * coef2);
}

// ---------------------------------------------------------------------------
extern "C" void kernel_launch(void* const* d_in, const int* in_sizes, int n_in,
                              void* d_out, int out_size, void* d_ws, size_t ws_size,
                              hipStream_t stream) {
    const float* x          = (const float*)d_in[0];
    const float* scores     = (const float*)d_in[1];
    const float* selfscores = (const float*)d_in[2];
    const int*   ctx_ids    = (const int*)d_in[3];
    const int*   prev_tok   = (const int*)d_in[4];
    const float* Wg         = (const float*)d_in[5];
    const float* bg         = (const float*)d_in[6];
    const float* W1         = (const float*)d_in[7];
    const float* b1         = (const float*)d_in[8];
    const float* W2         = (const float*)d_in[9];
    const float* b2         = (const float*)d_in[10];
    float* out = (float*)d_out;

    float* wsf    = (float*)d_ws;
    float* rowmax = wsf;          // 64
    float* rowsum = wsf + 64;     // 64
    float* mix    = wsf + 128;    // 64*3

    // 1) gate mix (independent, small)
    gate_kernel<<<BATCH, 128, 0, stream>>>(x, W1, b1, W2, b2, mix);

    // 2) main GEMM: logits -> d_out  (WMMA f32 16x16x4; 4 waves/block)
    const int gemm_blocks = (NTILES + 3) / 4;
    gen_gemm_kernel<<<gemm_blocks, 128, 0, stream>>>(x, Wg, bg, out);

    // 3) per-row softmax statistics
    rowstat_kernel<<<BATCH, 256, 0, stream>>>(out, rowmax, rowsum);

    // 4) normalize + weight by mix0 (in place)
    const int total = BATCH * VOCAB;
    norm_kernel<<<(total + 255) / 256, 256, 0, stream>>>(out, rowmax, rowsum, mix, total);

    // 5) scatter copy / self-copy mass
    scatter_kernel<<<BATCH, 128, 0, stream>>>(scores, selfscores, ctx_ids, prev_tok, mix, out);
}